// EdgeDistanceLoss_44117904064711
// MI455X (gfx1250) — compile-verified
//
#include <hip/hip_runtime.h>
#include <cstdint>

typedef float v2f __attribute__((ext_vector_type(2)));
typedef float v8f __attribute__((ext_vector_type(8)));
typedef unsigned int v4u __attribute__((ext_vector_type(4)));
typedef int v4i __attribute__((ext_vector_type(4)));
typedef int v8i __attribute__((ext_vector_type(8)));

#define IMG_H 1024
#define IMG_W 1024
#define TILE  64
#define RAW   72   // TILE + 2*4 halo (gauss 2 + sobel 1 + nms 1)
#define SMW   68   // smoothed region (TILE + 4)
#define MAGW  66   // magnitude region (TILE + 2)

// hbuf: 72 valid rows + 1 garbage row, 68 valid cols + 1 garbage col
#define HBP     69                         // hbuf pitch (garbage col = 68)
#define HBROWS  73                         // garbage row = 72
// smb: 68 valid rows/cols + 1 garbage row/col
#define SMBP    69                         // smb pitch (garbage col = 68)
#define SMBROWS 69                         // garbage row = 68

// flat LDS layout (floats)
#define RAW0_OFF  0                        // 72*72 raw pred tile, reused as mag pred
#define RAW1_OFF  (RAW0_OFF + RAW * RAW)   // 72*72 raw targ tile, reused as mag targ
#define HB_OFF    (RAW1_OFF + RAW * RAW)   // 73*69 row-smoothed (+garbage)
#define SMB_OFF   (HB_OFF + HBROWS * HBP)  // 69*69 fully smoothed (+garbage)
#define LDS_TOTAL (SMB_OFF + SMBROWS * SMBP)

__device__ __forceinline__ int imin(int a, int b) { return a < b ? a : b; }

// Banded separable-Gaussian weight: W[j][n] = g[j-n], g = {1,4,6,4,1}/16
__device__ __forceinline__ float bandw(int j, int n) {
  int d = j - n;
  float w = 0.0f;
  w = (d == 0 || d == 4) ? 0.0625f : w;
  w = (d == 1 || d == 3) ? 0.25f   : w;
  w = (d == 2)           ? 0.375f  : w;
  return w;
}

// Issue one TDM 2D tile load: 72x72 f32, row stride 1024 elements, into LDS.
// tensor_dims set large: interior tiles never go OOB (boundary tiles use the
// scalar path instead).
__device__ __forceinline__ void tdm_load_tile_f32(unsigned lds_byte_off, const float* gptr) {
  unsigned long long ga = (unsigned long long)(uintptr_t)gptr;
  v4u g0;
  g0.x = 1u;                                     // count = 1 valid descriptor
  g0.y = lds_byte_off;                           // LDS byte address
  g0.z = (unsigned)(ga & 0xFFFFFFFFu);           // global_addr[31:0]
  g0.w = (unsigned)((ga >> 32) & 0x1FFFFFFu)     // global_addr[56:32]
       | (2u << 30);                             // type = 2 ("image")
  const unsigned dim0 = 1u << 20, dim1 = 1u << 20;
  v8i g1;
  g1[0] = (int)(2u << 16);                       // wg_mask=0, data_size=2 (4B)
  g1[1] = (int)((dim0 & 0xFFFFu) << 16);         // tensor_dim0[15:0]
  g1[2] = (int)((dim0 >> 16) | ((dim1 & 0xFFFFu) << 16));
  g1[3] = (int)((dim1 >> 16) | ((unsigned)RAW << 16)); // tile_dim0 = 72
  g1[4] = (int)RAW;                              // tile_dim1 = 72, tile_dim2 = 0
  g1[5] = (int)IMG_W;                            // tensor_dim0_stride[31:0]
  g1[6] = 0;
  g1[7] = 0;
  v4i z4 = {0, 0, 0, 0};
  v8i z8 = {0, 0, 0, 0, 0, 0, 0, 0};
  __builtin_amdgcn_tensor_load_to_lds(g0, g1, z4, z4, z8, 0);
}

__global__ __launch_bounds__(256)
void canny_diff_kernel(const float* __restrict__ pred,
                       const float* __restrict__ targ,
                       unsigned* __restrict__ gcount) {
  __shared__ float lds[LDS_TOTAL];
  __shared__ unsigned blkCount;

  const int tid  = threadIdx.x;
  const int lane = tid & 31;
  const int wv   = tid >> 5;
  const int ln   = lane & 15;
  const int half = lane >> 4;
  const int kb   = half * 2;           // K-pair base per lane half (WMMA layout)

  const int tx = blockIdx.x, ty = blockIdx.y, bz = blockIdx.z;
  const int X0 = tx * TILE, Y0 = ty * TILE;
  const bool interior = (X0 >= 4) && (Y0 >= 4) &&
                        (X0 + TILE + 4 <= IMG_W) && (Y0 + TILE + 4 <= IMG_H);

  if (tid == 0) blkCount = 0;

  // Per-lane banded Gaussian weights; same VGPR striping serves as the
  // B-matrix of the row pass and the A-matrix of the column pass.
  v2f wB[5];
#pragma unroll
  for (int k = 0; k < 5; ++k) {
    wB[k].x = bandw(4 * k + kb, ln);
    wB[k].y = bandw(4 * k + kb + 1, ln);
  }

  const float* predB = pred + (size_t)bz * IMG_H * IMG_W;
  const float* targB = targ + (size_t)bz * IMG_H * IMG_W;

  // ---- prologue: stage BOTH 72x72 raw tiles; targ DMA overlaps pred compute
  if (interior) {
    if (wv == 0) {
      const float* src0 = predB + (Y0 - 4) * IMG_W + (X0 - 4);
      const float* src1 = targB + (Y0 - 4) * IMG_W + (X0 - 4);
      tdm_load_tile_f32((unsigned)(RAW0_OFF * 4), src0);
      tdm_load_tile_f32((unsigned)(RAW1_OFF * 4), src1);
    }
  } else {
    for (int idx = tid; idx < RAW * RAW; idx += 256) {
      int j = idx / RAW, i = idx - j * RAW;
      int gy = Y0 - 4 + j, gx = X0 - 4 + i;
      bool in = (gy >= 0 && gy < IMG_H && gx >= 0 && gx < IMG_W);
      int go = gy * IMG_W + gx;
      lds[RAW0_OFF + idx] = in ? predB[go] : 0.0f;
      lds[RAW1_OFF + idx] = in ? targB[go] : 0.0f;
    }
  }

  unsigned masks[2];

#pragma unroll
  for (int s = 0; s < 2; ++s) {
    const int rbase = (s == 0) ? RAW0_OFF : RAW1_OFF;

    if (interior && wv == 0) {
      if (s == 0) __builtin_amdgcn_s_wait_tensorcnt(1);  // pred tile landed
      else        __builtin_amdgcn_s_wait_tensorcnt(0);  // targ tile landed
    }
    __syncthreads();

    // ---- horizontal Gaussian: hbuf = raw(16x20) x Band(20x16), 25 blocks ----
    for (int bi = wv; bi < 25; bi += 8) {
      int r0 = (bi / 5) * 16, c0 = (bi % 5) * 16;
      v8f acc = {};
      int arow = imin(r0 + ln, RAW - 1);      // clamped rows store to garbage row 72
#pragma unroll
      for (int k = 0; k < 5; ++k) {
        int col = c0 + 4 * k + kb;
        v2f a;
        a.x = lds[rbase + arow * RAW + imin(col, RAW - 1)];
        a.y = lds[rbase + arow * RAW + imin(col + 1, RAW - 1)];
        acc = __builtin_amdgcn_wmma_f32_16x16x4_f32(
            false, a, false, wB[k], (short)0, acc, false, false);
      }
      // branchless stores: overhang rows/cols clamp into garbage row 72 /
      // garbage col 68 (never read)
      int ccst = imin(c0 + ln, SMW);          // 68 = garbage col
#pragma unroll
      for (int v = 0; v < 8; ++v) {
        int rrst = imin(r0 + v + half * 8, RAW);  // 72 = garbage row
        lds[HB_OFF + rrst * HBP + ccst] = acc[v];
      }
    }
    __syncthreads();

    // ---- vertical Gaussian: smb = Band(16x20) x hbuf(20x16) ----
    for (int bi = wv; bi < 25; bi += 8) {
      int r0 = (bi / 5) * 16, c0 = (bi % 5) * 16;
      v8f acc = {};
      int bcol = imin(c0 + ln, SMW - 1);
#pragma unroll
      for (int k = 0; k < 5; ++k) {
        v2f b;
        b.x = lds[HB_OFF + imin(r0 + 4 * k + kb,     RAW - 1) * HBP + bcol];
        b.y = lds[HB_OFF + imin(r0 + 4 * k + kb + 1, RAW - 1) * HBP + bcol];
        acc = __builtin_amdgcn_wmma_f32_16x16x4_f32(
            false, wB[k], false, b, (short)0, acc, false, false);
      }
      int ccst = imin(c0 + ln, SMW);          // 68 = garbage col
#pragma unroll
      for (int v = 0; v < 8; ++v) {
        int rrst = imin(r0 + v + half * 8, SMW);  // 68 = garbage row
        lds[SMB_OFF + rrst * SMBP + ccst] = acc[v];
      }
    }
    __syncthreads();

    // reference zero-pads the smoothed image for Sobel: zero OOB sm cells
    if (!interior) {
      for (int idx = tid; idx < SMW * SMW; idx += 256) {
        int j = idx / SMW, i = idx - j * SMW;
        int gy = Y0 - 2 + j, gx = X0 - 2 + i;
        if (gy < 0 || gy >= IMG_H || gx < 0 || gx >= IMG_W)
          lds[SMB_OFF + j * SMBP + i] = 0.0f;
      }
    }
    __syncthreads();

    // ---- Sobel + magnitude over 66x66 (overlays this image's raw tile) ----
    for (int idx = tid; idx < MAGW * MAGW; idx += 256) {
      int jm = idx / MAGW, im = idx - jm * MAGW;
      int ys = jm + 1, xs = im + 1;
#define S(dy, dx) lds[SMB_OFF + (ys + (dy)) * SMBP + (xs + (dx))]
      float gxv = (S(-1, 1) + 2.0f * S(0, 1) + S(1, 1))
                - (S(-1, -1) + 2.0f * S(0, -1) + S(1, -1));
      float gyv = (S(1, -1) + 2.0f * S(1, 0) + S(1, 1))
                - (S(-1, -1) + 2.0f * S(-1, 0) + S(-1, 1));
#undef S
      lds[rbase + idx] = sqrtf(gxv * gxv + gyv * gyv + 1e-12f);
    }
    __syncthreads();

    // ---- NMS + threshold: 16 center pixels per thread -> bitmask ----
    unsigned m = 0;
    int xc = tid & 63;
    int yb = (tid >> 6) * 16;
#pragma unroll 4
    for (int i = 0; i < 16; ++i) {
      int yc = yb + i;
      int ys = yc + 2, xs = xc + 2;
#define S(dy, dx) lds[SMB_OFF + (ys + (dy)) * SMBP + (xs + (dx))]
      float gxv = (S(-1, 1) + 2.0f * S(0, 1) + S(1, 1))
                - (S(-1, -1) + 2.0f * S(0, -1) + S(1, -1));
      float gyv = (S(1, -1) + 2.0f * S(1, 0) + S(1, 1))
                - (S(-1, -1) + 2.0f * S(-1, 0) + S(-1, 1));
#undef S
      // angle bin without atan2: reflect to gy>=0, compare against tan(22.5)
      float ax = gxv, ay = gyv;
      if (gyv < 0.0f) { ax = -ax; ay = -ay; }
      const float t = 0.41421356237309503f;
      float aax = fabsf(ax);
#define M(dy, dx) lds[rbase + (yc + 1 + (dy)) * MAGW + (xc + 1 + (dx))]
      float mc = M(0, 0), n1, n2;
      if (ay < t * aax)      { n1 = M(0, 1);  n2 = M(0, -1); }   // bin 0
      else if (aax < t * ay) { n1 = M(-1, 0); n2 = M(1, 0);  }   // bin 2
      else if (ax > 0.0f)    { n1 = M(1, 1);  n2 = M(-1, -1); }  // bin 1
      else                   { n1 = M(1, -1); n2 = M(-1, 1); }   // bin 3
#undef M
      bool keep = (mc >= n1) && (mc >= n2) && (mc > 0.5f);
      m |= (keep ? 1u : 0u) << i;
    }
    masks[s] = m;
    __syncthreads();   // next pass overwrites hbuf/smb
  }

  unsigned diff = __popc(masks[0] ^ masks[1]);
  atomicAdd(&blkCount, diff);
  __syncthreads();
  if (tid == 0) atomicAdd(gcount, blkCount);
}

__global__ void zero_count(unsigned* c) {
  if (threadIdx.x == 0) c[0] = 0u;
}

__global__ void finalize_loss(const unsigned* c, float* out) {
  if (threadIdx.x == 0)
    out[0] = (float)c[0] * (1.0f / (16.0f * 1024.0f * 1024.0f));
}

extern "C" void kernel_launch(void* const* d_in, const int* in_sizes, int n_in,
                              void* d_out, int out_size, void* d_ws, size_t ws_size,
                              hipStream_t stream) {
  (void)in_sizes; (void)n_in; (void)out_size; (void)ws_size;
  const float* pred = (const float*)d_in[0];
  const float* targ = (const float*)d_in[1];
  unsigned* cnt = (unsigned*)d_ws;

  zero_count<<<1, 32, 0, stream>>>(cnt);
  dim3 grid(IMG_W / TILE, IMG_H / TILE, 16);
  canny_diff_kernel<<<grid, 256, 0, stream>>>(pred, targ, cnt);
  finalize_loss<<<1, 32, 0, stream>>>(cnt, (float*)d_out);
}